// EdgeSheafLaplacian_81698867905234
// MI455X (gfx1250) — compile-verified
//
#include <hip/hip_runtime.h>
#include <cmath>

#define SHEAF_EPS 1e-4f

typedef float v2f_t __attribute__((ext_vector_type(2)));
typedef float v4f_t __attribute__((ext_vector_type(4)));
typedef float v8f_t __attribute__((ext_vector_type(8)));

// ---------------------------------------------------------------------------
// 8x8 fp32 matmul D = A * B via two chained V_WMMA_F32_16X16X4_F32.
// A, B, D are row-major 64-float arrays in LDS, private to the calling wave.
// Fragment layouts per CDNA5 ISA 7.12.2:
//   A (16x4, 2 VGPR): lane L -> m=L&15, VGPR0/1 = K=kb,kb+1 with kb=2*(L>>4)
//   B (4x16, 2 VGPR): lane L -> n=L&15, VGPR0/1 = K=kb,kb+1
//   C/D (16x16, 8 VGPR): VGPR r, lanes0-15 -> M=r (our 8x8 sits in r=0..7, n=0..7)
// ---------------------------------------------------------------------------
__device__ __forceinline__ void wmma8(const float* As, const float* Bs, float* Ds) {
  const int lane = threadIdx.x & 31;
  const int mn   = lane & 15;
  const int kb   = (lane >> 4) << 1;   // 0 or 2
  const bool ok  = mn < 8;             // rows/cols >= 8 are zero padding
  v2f_t a0, b0, a1, b1;
  a0.x = ok ? As[mn * 8 + kb]        : 0.0f;
  a0.y = ok ? As[mn * 8 + kb + 1]    : 0.0f;
  b0.x = ok ? Bs[kb * 8 + mn]        : 0.0f;
  b0.y = ok ? Bs[(kb + 1) * 8 + mn]  : 0.0f;
  a1.x = ok ? As[mn * 8 + 4 + kb]    : 0.0f;
  a1.y = ok ? As[mn * 8 + 5 + kb]    : 0.0f;
  b1.x = ok ? Bs[(4 + kb) * 8 + mn]  : 0.0f;
  b1.y = ok ? Bs[(5 + kb) * 8 + mn]  : 0.0f;
  v8f_t c = {};
  c = __builtin_amdgcn_wmma_f32_16x16x4_f32(false, a0, false, b0, (short)0, c, false, false);
  c = __builtin_amdgcn_wmma_f32_16x16x4_f32(false, a1, false, b1, (short)0, c, false, false);
  if (lane < 8) {
#pragma unroll
    for (int r = 0; r < 8; ++r) Ds[r * 8 + lane] = c[r];
  }
}

// ---------------------------------------------------------------------------
// K0: zero the 256 MB output with non-temporal float4 stores (output > L2).
// ---------------------------------------------------------------------------
__global__ void zero_out_kernel(v4f_t* __restrict__ out, long long n4) {
  long long i      = (long long)blockIdx.x * blockDim.x + threadIdx.x;
  long long stride = (long long)gridDim.x * blockDim.x;
  v4f_t z = {0.0f, 0.0f, 0.0f, 0.0f};
  for (; i < n4; i += stride) __builtin_nontemporal_store(z, &out[i]);
}

// K1a: reverse-edge map <- -1
__global__ void init_map_kernel(int* __restrict__ map, long long nn) {
  long long i      = (long long)blockIdx.x * blockDim.x + threadIdx.x;
  long long stride = (long long)gridDim.x * blockDim.x;
  for (; i < nn; i += stride) map[i] = -1;
}

// K1b: diag accumulator <- 0
__global__ void zero_f32_kernel(float* __restrict__ p, long long cnt) {
  long long i      = (long long)blockIdx.x * blockDim.x + threadIdx.x;
  long long stride = (long long)gridDim.x * blockDim.x;
  for (; i < cnt; i += stride) p[i] = 0.0f;
}

// K2: scatter map[i*n+j] = p  (directed edges are unique)
__global__ void scatter_map_kernel(const int* __restrict__ ei, int* __restrict__ map,
                                   int P, int n) {
  int p = blockIdx.x * blockDim.x + threadIdx.x;
  if (p >= P) return;
  int i = ei[2 * p], j = ei[2 * p + 1];
  map[(long long)i * n + j] = p;
}

// ---------------------------------------------------------------------------
// K3: per edge p (one wave): FtF_p = R_p^T R_p via WMMA, atomic segment-sum
// into diag[head(p)].
// ---------------------------------------------------------------------------
__global__ __launch_bounds__(128) void ftf_kernel(const float* __restrict__ R,
                                                  const int* __restrict__ ei,
                                                  float* __restrict__ diag, int P) {
  __shared__ float sA[4][64], sB[4][64], sD[4][64];
  const int w = threadIdx.x >> 5, lane = threadIdx.x & 31;
  const int p = blockIdx.x * 4 + w;
  if (p >= P) return;                      // wave-uniform exit
  const float* Rp = R + (size_t)p * 64;
  float x0 = Rp[lane], x1 = Rp[lane + 32];
  sB[w][lane]      = x0;                                       // B = R_p
  sB[w][lane + 32] = x1;
  sA[w][((lane & 7) << 3) | (lane >> 3)]        = x0;          // A = R_p^T
  sA[w][((lane & 7) << 3) | ((lane >> 3) + 4)]  = x1;
  wmma8(sA[w], sB[w], sD[w]);                                  // D = R_p^T R_p
  const int i = ei[2 * p];
  float* dg = diag + (size_t)i * 64;
  atomicAdd(&dg[lane],      sD[w][lane]);
  atomicAdd(&dg[lane + 32], sD[w][lane + 32]);
}

// ---------------------------------------------------------------------------
// K4: per node v (one thread): D = sym(diag_v) + eps*I ; Jacobi eigensolve ;
// Dis_v = V diag(1/sqrt(max(w,eps))) V^T.
// ---------------------------------------------------------------------------
__global__ void dis_kernel(const float* __restrict__ diag, float* __restrict__ Dis, int n) {
  int v = blockIdx.x * blockDim.x + threadIdx.x;
  if (v >= n) return;
  const float* d = diag + (size_t)v * 64;
  float A[64], V[64];
#pragma unroll
  for (int i = 0; i < 8; ++i)
#pragma unroll
    for (int j = 0; j < 8; ++j) {
      A[i * 8 + j] = 0.5f * (d[i * 8 + j] + d[j * 8 + i]) + (i == j ? SHEAF_EPS : 0.0f);
      V[i * 8 + j] = (i == j) ? 1.0f : 0.0f;
    }
  for (int sweep = 0; sweep < 12; ++sweep) {
    for (int p = 0; p < 7; ++p) {
      for (int q = p + 1; q < 8; ++q) {
        float apq = A[p * 8 + q];
        float c, s;
        if (fabsf(apq) > 1e-12f) {
          float tau = (A[q * 8 + q] - A[p * 8 + p]) / (2.0f * apq);
          float t   = copysignf(1.0f, tau) / (fabsf(tau) + sqrtf(tau * tau + 1.0f));
          c = 1.0f / sqrtf(t * t + 1.0f);
          s = t * c;
        } else { c = 1.0f; s = 0.0f; }
#pragma unroll
        for (int k = 0; k < 8; ++k) {          // A <- A G
          float akp = A[k * 8 + p], akq = A[k * 8 + q];
          A[k * 8 + p] = c * akp - s * akq;
          A[k * 8 + q] = s * akp + c * akq;
        }
#pragma unroll
        for (int k = 0; k < 8; ++k) {          // A <- G^T A
          float apk = A[p * 8 + k], aqk = A[q * 8 + k];
          A[p * 8 + k] = c * apk - s * aqk;
          A[q * 8 + k] = s * apk + c * aqk;
        }
#pragma unroll
        for (int k = 0; k < 8; ++k) {          // V <- V G
          float vkp = V[k * 8 + p], vkq = V[k * 8 + q];
          V[k * 8 + p] = c * vkp - s * vkq;
          V[k * 8 + q] = s * vkp + c * vkq;
        }
      }
    }
  }
  float wi[8];
#pragma unroll
  for (int l = 0; l < 8; ++l) {
    float wl = A[l * 8 + l];
    wl = wl < SHEAF_EPS ? SHEAF_EPS : wl;
    wi[l] = 1.0f / sqrtf(wl);
  }
  float* o = Dis + (size_t)v * 64;
#pragma unroll
  for (int i = 0; i < 8; ++i)
#pragma unroll
    for (int j = 0; j < 8; ++j) {
      float acc = 0.0f;
#pragma unroll
      for (int l = 0; l < 8; ++l) acc += V[i * 8 + l] * wi[l] * V[j * 8 + l];
      o[i * 8 + j] = acc;
    }
}

// ---------------------------------------------------------------------------
// K5: per node v (one wave): out block (v,v) = Dis_v * diag_v * Dis_v (2 WMMAs)
// ---------------------------------------------------------------------------
__global__ __launch_bounds__(128) void diagblock_kernel(const float* __restrict__ diag,
                                                        const float* __restrict__ Dis,
                                                        float* __restrict__ out, int n) {
  __shared__ float sA[4][64], sB[4][64], sC[4][64];
  const int w = threadIdx.x >> 5, lane = threadIdx.x & 31;
  const int v = blockIdx.x * 4 + w;
  if (v >= n) return;
  const float* dv = diag + (size_t)v * 64;
  const float* sv = Dis  + (size_t)v * 64;
  sA[w][lane] = sv[lane]; sA[w][lane + 32] = sv[lane + 32];   // A = Dis_v
  sB[w][lane] = dv[lane]; sB[w][lane + 32] = dv[lane + 32];   // B = diag_v
  wmma8(sA[w], sB[w], sC[w]);                                 // C = Dis_v * diag_v
  wmma8(sC[w], sA[w], sB[w]);                                 // B = C * Dis_v
  const size_t ld = (size_t)n * 8;
  const int e = lane * 2, r = e >> 3, cidx = e & 7;
  float2 val = make_float2(sB[w][e], sB[w][e + 1]);
  *(float2*)(out + ((size_t)v * 8 + r) * ld + (size_t)v * 8 + cidx) = val;
}

// ---------------------------------------------------------------------------
// K6: per edge p (one wave): out block (i,j) =
//     Dis_i * (-sign(L1[i,j]) * R_q^T R_p) * Dis_j   (3 WMMAs)
// ---------------------------------------------------------------------------
__global__ __launch_bounds__(128) void offblock_kernel(const float* __restrict__ R,
                                                       const int* __restrict__ ei,
                                                       const int* __restrict__ map,
                                                       const float* __restrict__ L1,
                                                       const float* __restrict__ Dis,
                                                       float* __restrict__ out,
                                                       int P, int n) {
  __shared__ float sA[4][64], sB[4][64], sC[4][64];
  const int w = threadIdx.x >> 5, lane = threadIdx.x & 31;
  const int p = blockIdx.x * 4 + w;
  if (p >= P) return;                       // wave-uniform
  const int i = ei[2 * p], j = ei[2 * p + 1];
  const int q = map[(long long)j * n + i];
  if (q < 0) return;                        // reverse edge absent -> block stays 0
  const float l1 = L1[(long long)i * n + j];
  const float scale = (l1 > 0.0f) ? -1.0f : (l1 < 0.0f ? 1.0f : 0.0f);  // -sign
  const float* Rp = R + (size_t)p * 64;
  const float* Rq = R + (size_t)q * 64;
  float q0 = Rq[lane], q1 = Rq[lane + 32];
  sA[w][((lane & 7) << 3) | (lane >> 3)]       = scale * q0;  // A = -sgn * R_q^T
  sA[w][((lane & 7) << 3) | ((lane >> 3) + 4)] = scale * q1;
  sB[w][lane]      = Rp[lane];                                // B = R_p
  sB[w][lane + 32] = Rp[lane + 32];
  wmma8(sA[w], sB[w], sC[w]);                                 // C = -sgn * R_q^T R_p
  const float* di = Dis + (size_t)i * 64;
  sA[w][lane] = di[lane]; sA[w][lane + 32] = di[lane + 32];
  wmma8(sA[w], sC[w], sB[w]);                                 // B = Dis_i * C
  const float* dj = Dis + (size_t)j * 64;
  sA[w][lane] = dj[lane]; sA[w][lane + 32] = dj[lane + 32];
  wmma8(sB[w], sA[w], sC[w]);                                 // C = B * Dis_j
  const size_t ld = (size_t)n * 8;
  const int e = lane * 2, r = e >> 3, cidx = e & 7;
  float2 val = make_float2(sC[w][e], sC[w][e + 1]);
  *(float2*)(out + ((size_t)i * 8 + r) * ld + (size_t)j * 8 + cidx) = val;
}

// ---------------------------------------------------------------------------
extern "C" void kernel_launch(void* const* d_in, const int* in_sizes, int n_in,
                              void* d_out, int out_size, void* d_ws, size_t ws_size,
                              hipStream_t stream) {
  const float* R  = (const float*)d_in[0];   // (P, 8, 8) fp32
  const int*   ei = (const int*)d_in[1];     // (P, 2)    int32
  const float* L1 = (const float*)d_in[3];   // (n, n)    fp32

  const int P = in_sizes[1] / 2;
  const int n = (int)(sqrt((double)in_sizes[3]) + 0.5);  // L1 is n*n

  // Workspace layout: [revmap: n*n int32][diag: n*64 f32][Dis: n*64 f32]
  int*   map  = (int*)d_ws;
  float* diag = (float*)((char*)d_ws + (size_t)n * n * sizeof(int));
  float* Dis  = diag + (size_t)n * 64;

  float* out = (float*)d_out;

  // K0: zero 256 MB output (NT stores)
  const long long n4 = (long long)out_size / 4;
  zero_out_kernel<<<8192, 256, 0, stream>>>((v4f_t*)out, n4);

  // K1: init map and diag accumulator
  init_map_kernel<<<1024, 256, 0, stream>>>(map, (long long)n * n);
  zero_f32_kernel<<<256, 256, 0, stream>>>(diag, (long long)n * 64);

  // K2: scatter reverse-edge map
  scatter_map_kernel<<<(P + 255) / 256, 256, 0, stream>>>(ei, map, P, n);

  // K3: FtF + segment-sum into diag
  ftf_kernel<<<(P + 3) / 4, 128, 0, stream>>>(R, ei, diag, P);

  // K4: per-node D^{-1/2}
  dis_kernel<<<(n + 127) / 128, 128, 0, stream>>>(diag, Dis, n);

  // K5: diagonal blocks
  diagblock_kernel<<<(n + 3) / 4, 128, 0, stream>>>(diag, Dis, out, n);

  // K6: off-diagonal edge blocks
  offblock_kernel<<<(P + 3) / 4, 128, 0, stream>>>(R, ei, map, L1, Dis, out, P, n);
}